// TaskAlignedAssigner_8177617731725
// MI455X (gfx1250) — compile-verified
//
#include <hip/hip_runtime.h>
#include <hip/hip_bf16.h>
#include <stdint.h>

#define B_    32
#define T_    32
#define A_    8400
#define C_    80
#define TOPK_ 13
#define EPS_  1e-9f
#define CEPS_ 1e-7f
#define CHUNK 512

typedef uint32_t u32;

// ---------------- CDNA5 TDM path (guarded; fallback = plain LDS fill) -------
#if defined(__gfx1250__) && __has_builtin(__builtin_amdgcn_tensor_load_to_lds) && __has_builtin(__builtin_amdgcn_s_wait_tensorcnt)
#define USE_TDM 1
typedef unsigned int v4u __attribute__((ext_vector_type(4)));
typedef int          v4i __attribute__((ext_vector_type(4)));
typedef int          v8i __attribute__((ext_vector_type(8)));

__device__ __forceinline__ u32 lds_off32(const void* p) {
  // generic LDS address: low 32 bits are the allocation-relative LDS offset
  return (u32)(uintptr_t)p;
}

// 1-D copy of nelem4 dwords from global to LDS via the Tensor Data Mover.
// D# layout per CDNA5 ISA (group0 + group1, groups 2/3 zero).
__device__ __forceinline__ void tdm_load_1d_f32(u32 lds_byte, const void* gsrc, u32 nelem4) {
  uint64_t ga = (uint64_t)(uintptr_t)gsrc;
  v4u g0;
  g0.x = 1u;                                  // count=1, user descriptor
  g0.y = lds_byte;                            // lds_addr
  g0.z = (u32)ga;                             // global_addr[31:0]
  g0.w = (u32)(ga >> 32) | (2u << 30);        // global_addr[56:32] | type=2
  v8i g1;
  g1[0] = (int)(2u << 16);                    // data_size = 2 (4 bytes)
  g1[1] = (int)((nelem4 & 0xFFFFu) << 16);    // tensor_dim0[15:0] -> bits[63:48]
  g1[2] = (int)((nelem4 >> 16) | (1u << 16)); // tensor_dim0 hi | tensor_dim1=1
  g1[3] = (int)(nelem4 << 16);                // tile_dim0 -> bits[127:112]
  g1[4] = 0;                                  // tile_dim1=0, tile_dim2=0
  g1[5] = (int)nelem4;                        // tensor_dim0_stride (low)
  g1[6] = 0;
  g1[7] = 0;
  v4i z4 = {0, 0, 0, 0};
  v8i z8 = {0, 0, 0, 0, 0, 0, 0, 0};
  // 6-arg toolchain form: (g0, g1, g2, g3, g4, cpol)
  __builtin_amdgcn_tensor_load_to_lds(g0, g1, z4, z4, z8, 0);
}
#endif

// ------------------------------- Kernel 1 -----------------------------------
// fuse[b,t,a] and iou[b,t,a].  One block = (b, 512-anchor chunk); TDM stages
// the chunk's pd_bboxes + anchor points into LDS, reused across all 32 t.
__global__ __launch_bounds__(256) void k1_fuse_iou(
    const float* __restrict__ anchors, const float* __restrict__ pd_class,
    const float* __restrict__ pd_bboxes, const int* __restrict__ gt_class,
    const float* __restrict__ gt_bboxes, const float* __restrict__ mask_gt,
    float* __restrict__ fuseW, float* __restrict__ iouW) {
  __shared__ float s_pd[CHUNK * 4];
  __shared__ float s_anc[CHUNK * 2];
  __shared__ float s_gtb[T_ * 4];
  __shared__ float s_mgt[T_];
  __shared__ int   s_gtc[T_];

  const int b   = blockIdx.y;
  const int a0  = blockIdx.x * CHUNK;
  const int n   = min(CHUNK, A_ - a0);
  const int tid = threadIdx.x;

  if (tid == 0)
    __builtin_prefetch(&pd_class[((size_t)b * A_ + a0) * C_], 0, 1);

#if defined(USE_TDM)
  if (tid < 32) {  // one wave issues the DMA (descriptor is wave-uniform)
    tdm_load_1d_f32(lds_off32(s_pd),  pd_bboxes + ((size_t)b * A_ + a0) * 4, (u32)(n * 4));
    tdm_load_1d_f32(lds_off32(s_anc), anchors + (size_t)a0 * 2,              (u32)(n * 2));
    __builtin_amdgcn_s_wait_tensorcnt(0);
  }
#else
  for (int i = tid; i < n * 4; i += 256) s_pd[i]  = pd_bboxes[((size_t)b * A_ + a0) * 4 + i];
  for (int i = tid; i < n * 2; i += 256) s_anc[i] = anchors[(size_t)a0 * 2 + i];
#endif
  if (tid < T_) {
    s_gtb[tid * 4 + 0] = gt_bboxes[((size_t)b * T_ + tid) * 4 + 0];
    s_gtb[tid * 4 + 1] = gt_bboxes[((size_t)b * T_ + tid) * 4 + 1];
    s_gtb[tid * 4 + 2] = gt_bboxes[((size_t)b * T_ + tid) * 4 + 2];
    s_gtb[tid * 4 + 3] = gt_bboxes[((size_t)b * T_ + tid) * 4 + 3];
    s_mgt[tid] = mask_gt[b * T_ + tid];
    s_gtc[tid] = gt_class[b * T_ + tid];
  }
  __syncthreads();

  const float k4pi2 = 4.0f / (3.14159265358979f * 3.14159265358979f);
  for (int t = 0; t < T_; ++t) {
    const float gx1 = s_gtb[t * 4 + 0], gy1 = s_gtb[t * 4 + 1];
    const float gx2 = s_gtb[t * 4 + 2], gy2 = s_gtb[t * 4 + 3];
    const float mg  = s_mgt[t];
    const int   gc  = s_gtc[t];
    const float w1 = gx2 - gx1, h1 = gy2 - gy1 + CEPS_;
    const float at1 = atanf(w1 / h1);
    for (int i = tid; i < n; i += 256) {
      const int a = a0 + i;
      const float ax = s_anc[i * 2 + 0], ay = s_anc[i * 2 + 1];
      const float dmin = fminf(fminf(ax - gx1, ay - gy1), fminf(gx2 - ax, gy2 - ay));
      const float m = (dmin > EPS_) ? mg : 0.0f;
      float iou = 0.0f, fu = 0.0f;
      if (m > 0.0f) {
        const float px1 = s_pd[i * 4 + 0], py1 = s_pd[i * 4 + 1];
        const float px2 = s_pd[i * 4 + 2], py2 = s_pd[i * 4 + 3];
        const float w2 = px2 - px1, h2 = py2 - py1 + CEPS_;
        float iw = fminf(gx2, px2) - fmaxf(gx1, px1); iw = fmaxf(iw, 0.0f);
        float ih = fminf(gy2, py2) - fmaxf(gy1, py1); ih = fmaxf(ih, 0.0f);
        const float inter = iw * ih;
        const float uni   = w1 * h1 + w2 * h2 - inter + CEPS_;
        const float iou0  = inter / uni;
        const float cw = fmaxf(gx2, px2) - fminf(gx1, px1);
        const float ch = fmaxf(gy2, py2) - fminf(gy1, py1);
        const float c2 = cw * cw + ch * ch + CEPS_;
        const float dx = px1 + px2 - gx1 - gx2, dy = py1 + py2 - gy1 - gy2;
        const float rho2 = (dx * dx + dy * dy) * 0.25f;
        const float dat = atanf(w2 / h2) - at1;
        const float v = k4pi2 * dat * dat;
        const float alpha = v / (v - iou0 + (1.0f + CEPS_));
        const float ciou = iou0 - (rho2 / c2 + v * alpha);
        iou = fmaxf(ciou, 0.0f) * m;
        const float score = pd_class[((size_t)b * A_ + a) * C_ + gc] * m;
        const float i2 = iou * iou;
        fu = score * i2 * i2 * i2;
      }
      const size_t o = ((size_t)(b * T_ + t)) * A_ + a;
      fuseW[o] = fu;
      iouW[o]  = iou;
    }
  }
}

// ------------------------------- Kernel 2 -----------------------------------
// Top-13 per (b,t) row; ties -> smaller index (matches lax.top_k stability).
__global__ __launch_bounds__(256) void k2_topk(
    const float* __restrict__ fuseW, int* __restrict__ topkW,
    float* __restrict__ posAlign, float* __restrict__ posIou) {
  __shared__ float lv[256 * TOPK_];
  __shared__ int   li[256 * TOPK_];
  __shared__ float rv[256];
  __shared__ int   ri[256];
  __shared__ int   ro[256];
  const int row = blockIdx.x;   // b*T + t
  const int tid = threadIdx.x;
  float* mylv = &lv[tid * TOPK_];
  int*   myli = &li[tid * TOPK_];
  for (int j = 0; j < TOPK_; ++j) { mylv[j] = -1.0f; myli[j] = 0x7FFFFFFF; }

  const float* frow = fuseW + (size_t)row * A_;
  for (int a = tid; a < A_; a += 256) {
    const float v = frow[a];
    if (v > mylv[TOPK_ - 1]) {
      int j = TOPK_ - 1;
      while (j > 0 && v > mylv[j - 1]) { mylv[j] = mylv[j - 1]; myli[j] = myli[j - 1]; --j; }
      mylv[j] = v; myli[j] = a;
    }
  }
  __syncthreads();

  int p = 0;
  for (int r = 0; r < TOPK_; ++r) {
    rv[tid] = (p < TOPK_) ? mylv[p] : -2.0f;
    ri[tid] = (p < TOPK_) ? myli[p] : 0x7FFFFFFF;
    ro[tid] = tid;
    __syncthreads();
    for (int s = 128; s > 0; s >>= 1) {
      if (tid < s) {
        const float v2 = rv[tid + s]; const int i2 = ri[tid + s];
        if (v2 > rv[tid] || (v2 == rv[tid] && i2 < ri[tid])) {
          rv[tid] = v2; ri[tid] = i2; ro[tid] = ro[tid + s];
        }
      }
      __syncthreads();
    }
    const int owner = ro[0];
    if (tid == 0) topkW[row * TOPK_ + r] = ri[0];
    __syncthreads();
    if (tid == owner) ++p;
  }
  if (tid == 0) { posAlign[row] = 0.0f; posIou[row] = 0.0f; }
}

// ------------------------------- Kernel 3 -----------------------------------
// Per-anchor assignment resolution + pos_align/pos_iou accumulation.
__global__ __launch_bounds__(256) void k3_resolve(
    const float* __restrict__ anchors, const float* __restrict__ gt_bboxes,
    const float* __restrict__ mask_gt, const int* __restrict__ topkW,
    const float* __restrict__ fuseW, const float* __restrict__ iouW,
    int* __restrict__ tAssigned, float* __restrict__ posAlign,
    float* __restrict__ posIou, float* __restrict__ outFg,
    float* __restrict__ outTgt) {
  __shared__ int   s_tk[T_ * TOPK_];
  __shared__ float s_gtb[T_ * 4];
  __shared__ float s_mgt[T_];
  const int b   = blockIdx.y;
  const int tid = threadIdx.x;
  const int a   = blockIdx.x * 256 + tid;
  for (int i = tid; i < T_ * TOPK_; i += 256) s_tk[i] = topkW[b * T_ * TOPK_ + i];
  if (tid < T_) {
    s_gtb[tid * 4 + 0] = gt_bboxes[((size_t)b * T_ + tid) * 4 + 0];
    s_gtb[tid * 4 + 1] = gt_bboxes[((size_t)b * T_ + tid) * 4 + 1];
    s_gtb[tid * 4 + 2] = gt_bboxes[((size_t)b * T_ + tid) * 4 + 2];
    s_gtb[tid * 4 + 3] = gt_bboxes[((size_t)b * T_ + tid) * 4 + 3];
    s_mgt[tid] = mask_gt[b * T_ + tid];
  }
  __syncthreads();
  if (a >= A_) return;

  const float ax = anchors[a * 2 + 0], ay = anchors[a * 2 + 1];
  int cnt = 0, tSel = -1;
  for (int t = 0; t < T_; ++t) {
    bool intopk = false;
    for (int j = 0; j < TOPK_; ++j) intopk |= (s_tk[t * TOPK_ + j] == a);
    if (!intopk) continue;
    const float dmin = fminf(fminf(ax - s_gtb[t * 4 + 0], ay - s_gtb[t * 4 + 1]),
                             fminf(s_gtb[t * 4 + 2] - ax, s_gtb[t * 4 + 3] - ay));
    if (dmin > EPS_ && s_mgt[t] > 0.0f) { if (cnt == 0) tSel = t; ++cnt; }
  }
  int assigned;
  if (cnt > 1) {                      // multi-GT anchor: argmax iou over t (first max)
    float best = -1.0f; int bt = 0;
    for (int t = 0; t < T_; ++t) {
      const float v = iouW[((size_t)(b * T_ + t)) * A_ + a];
      if (v > best) { best = v; bt = t; }
    }
    assigned = bt;
  } else if (cnt == 1) assigned = tSel;
  else assigned = -1;

  tAssigned[b * A_ + a] = assigned;
  outFg[b * A_ + a]  = (assigned >= 0) ? 1.0f : 0.0f;
  outTgt[b * A_ + a] = (float)((assigned >= 0) ? assigned : 0);
  if (assigned >= 0) {
    const size_t o = ((size_t)(b * T_ + assigned)) * A_ + a;
    atomicMax((int*)&posAlign[b * T_ + assigned], __float_as_int(fuseW[o]));
    atomicMax((int*)&posIou[b * T_ + assigned],  __float_as_int(iouW[o]));
  }
}

// ------------------------------- Kernel 4 -----------------------------------
// Gather targets, compute norm, write one-hot*norm rows with float4 stores.
__global__ __launch_bounds__(256) void k4_outputs(
    const int* __restrict__ gt_class, const float* __restrict__ gt_bboxes,
    const int* __restrict__ tAssigned, const float* __restrict__ fuseW,
    const float* __restrict__ posAlign, const float* __restrict__ posIou,
    float* __restrict__ outCls, float* __restrict__ outBox,
    float* __restrict__ outMask) {
  const int b = blockIdx.y;
  const int a = blockIdx.x * 256 + threadIdx.x;
  if (a >= A_) return;
  const int assigned = tAssigned[b * A_ + a];
  const int tEff = (assigned >= 0) ? assigned : 0;
  int cls = gt_class[b * T_ + tEff]; if (cls < 0) cls = 0;
  outCls[b * A_ + a] = (float)cls;
  const float* gb = gt_bboxes + ((size_t)(b * T_ + tEff)) * 4;
  float4 bx; bx.x = gb[0]; bx.y = gb[1]; bx.z = gb[2]; bx.w = gb[3];
  ((float4*)outBox)[(size_t)b * A_ + a] = bx;
  float norm = 0.0f;
  if (assigned >= 0) {
    const size_t o = ((size_t)(b * T_ + assigned)) * A_ + a;
    norm = fuseW[o] * posIou[b * T_ + assigned] / (posAlign[b * T_ + assigned] + EPS_);
  }
  float* mrow = outMask + ((size_t)(b * A_ + a)) * C_;
  for (int j = 0; j < C_; j += 4) {
    float4 z; z.x = 0.0f; z.y = 0.0f; z.z = 0.0f; z.w = 0.0f;
    if (assigned >= 0 && cls >= j && cls < j + 4) ((float*)&z)[cls - j] = norm;
    ((float4*)(mrow + j))[0] = z;
  }
}

// ------------------------------- Launch -------------------------------------
extern "C" void kernel_launch(void* const* d_in, const int* in_sizes, int n_in,
                              void* d_out, int out_size, void* d_ws, size_t ws_size,
                              hipStream_t stream) {
  const float* anchors   = (const float*)d_in[0];
  const float* pd_class  = (const float*)d_in[1];
  const float* pd_bboxes = (const float*)d_in[2];
  const int*   gt_class  = (const int*)d_in[3];
  const float* gt_bboxes = (const float*)d_in[4];
  const float* mask_gt   = (const float*)d_in[5];

  char* w = (char*)d_ws;
  float* fuseW    = (float*)w; w += (size_t)B_ * T_ * A_ * sizeof(float);
  float* iouW     = (float*)w; w += (size_t)B_ * T_ * A_ * sizeof(float);
  int*   topkW    = (int*)w;   w += (size_t)B_ * T_ * TOPK_ * sizeof(int);
  float* posAlign = (float*)w; w += (size_t)B_ * T_ * sizeof(float);
  float* posIou   = (float*)w; w += (size_t)B_ * T_ * sizeof(float);
  int*   tAssigned= (int*)w;   w += (size_t)B_ * A_ * sizeof(int);

  float* out     = (float*)d_out;
  float* outCls  = out;
  float* outBox  = outCls + (size_t)B_ * A_;
  float* outMask = outBox + (size_t)B_ * A_ * 4;
  float* outFg   = outMask + (size_t)B_ * A_ * C_;
  float* outTgt  = outFg + (size_t)B_ * A_;

  dim3 blk(256);
  dim3 g1((A_ + CHUNK - 1) / CHUNK, B_);
  k1_fuse_iou<<<g1, blk, 0, stream>>>(anchors, pd_class, pd_bboxes, gt_class,
                                      gt_bboxes, mask_gt, fuseW, iouW);
  k2_topk<<<dim3(B_ * T_), blk, 0, stream>>>(fuseW, topkW, posAlign, posIou);
  dim3 g3((A_ + 255) / 256, B_);
  k3_resolve<<<g3, blk, 0, stream>>>(anchors, gt_bboxes, mask_gt, topkW, fuseW,
                                     iouW, tAssigned, posAlign, posIou, outFg, outTgt);
  k4_outputs<<<g3, blk, 0, stream>>>(gt_class, gt_bboxes, tAssigned, fuseW,
                                     posAlign, posIou, outCls, outBox, outMask);
}